// D_TopoEncoder_67473936221017
// MI455X (gfx1250) — compile-verified
//
#include <hip/hip_runtime.h>
#include <hip/hip_bf16.h>
#include <math.h>

typedef __attribute__((ext_vector_type(2))) float v2f;
typedef __attribute__((ext_vector_type(8))) float v8f;

#define NB 512   // batch (N*M)
#define CC 3     // channels
#define TT 128   // time
#define VV 25    // joints
#define EE 64    // structure elements
#define PP (VV - 1)  // persistence pairs per batch

// ---------------------------------------------------------------------------
// Kernel 0: reset global min/max (bit-monotone uint atomics; all dists > 0)
// ---------------------------------------------------------------------------
__global__ void init_minmax_kernel(unsigned* mm) {
  if (threadIdx.x == 0) {
    mm[0] = 0x7F800000u;  // +inf bits (min accumulator)
    mm[1] = 0u;           // 0.0 bits  (max accumulator)
  }
}

// ---------------------------------------------------------------------------
// Kernel 1: T-mean  xm[b,c,v] = mean_t x[b,c,t,v]
// ---------------------------------------------------------------------------
__global__ void mean_t_kernel(const float* __restrict__ x,
                              float* __restrict__ xm) {
  const int b = blockIdx.x;
  const int tid = threadIdx.x;
  if (tid >= CC * VV) return;
  const int c = tid / VV;
  const int v = tid % VV;
  const float* p = x + ((size_t)(b * CC + c) * TT) * VV + v;
  float s = 0.0f;
#pragma unroll 4
  for (int t = 0; t < TT; ++t) s += p[t * VV];
  xm[(b * CC + c) * VV + v] = s * (1.0f / TT);
}

// ---------------------------------------------------------------------------
// Kernel 2: per-batch (one wave32): Gram via WMMA -> 25x25 distance matrix in
// LDS -> global min/max atomics -> Prim MST deaths (raw, unnormalized).
// ---------------------------------------------------------------------------
__global__ void dist_mst_kernel(const float* __restrict__ xm,
                                float* __restrict__ deaths,
                                unsigned* __restrict__ mm) {
  const int b = blockIdx.x;
  const int lane = threadIdx.x;  // 0..31, one wave

  __shared__ float Dl[VV * 32];  // distance matrix, row stride 32
  __shared__ float n2[32];       // squared norms per joint

  const float* xb = xm + b * CC * VV;

  // Squared norm per joint v == lane
  float nn = 0.0f;
  if (lane < VV) {
    const float a0 = xb[0 * VV + lane];
    const float a1 = xb[1 * VV + lane];
    const float a2 = xb[2 * VV + lane];
    nn = a0 * a0 + a1 * a1 + a2 * a2;
  }
  n2[lane] = nn;

  // WMMA fragments. 32-bit 16x4 A layout (ISA 7.12.2): lane&15 = matrix row,
  // lane>>4 selects K pair {0,1} vs {2,3}; VGPR0=K_even, VGPR1=K_odd.
  // B (4x16) mirrors it with lane&15 = column. For a Gram matrix both
  // fragments are built with the identical per-lane formula.
  const int khalf = lane >> 4;
  const int idx = lane & 15;
  const int c0 = 2 * khalf;      // 0 or 2
  const int c1 = 2 * khalf + 1;  // 1 or 3 (3 => zero pad)

  v2f fr0, fr1;
  fr0.x = (c0 < CC) ? xb[c0 * VV + idx] : 0.0f;   // v = idx (<16)
  fr0.y = (c1 < CC) ? xb[c1 * VV + idx] : 0.0f;
  const int vhi = 16 + idx;                        // v = 16..31 (pad >= 25)
  fr0 = fr0;  // keep simple
  fr1.x = (c0 < CC && vhi < VV) ? xb[c0 * VV + vhi] : 0.0f;
  fr1.y = (c1 < CC && vhi < VV) ? xb[c1 * VV + vhi] : 0.0f;

  const v8f z = {};
  v8f g00 = __builtin_amdgcn_wmma_f32_16x16x4_f32(false, fr0, false, fr0,
                                                  (short)0, z, false, false);
  v8f g01 = __builtin_amdgcn_wmma_f32_16x16x4_f32(false, fr0, false, fr1,
                                                  (short)0, z, false, false);
  v8f g10 = __builtin_amdgcn_wmma_f32_16x16x4_f32(false, fr1, false, fr0,
                                                  (short)0, z, false, false);
  v8f g11 = __builtin_amdgcn_wmma_f32_16x16x4_f32(false, fr1, false, fr1,
                                                  (short)0, z, false, false);

  __syncthreads();  // n2[] visible

  // C/D 16x16 f32 layout: VGPR r, lane L -> M = r + 8*(L>>4), N = L&15.
  const int col = idx;
  const int rbase = 8 * khalf;

  auto store_tile = [&](const v8f& g, int mt, int nt) {
#pragma unroll
    for (int r = 0; r < 8; ++r) {
      const int row = mt * 16 + rbase + r;
      const int cv = nt * 16 + col;
      if (row < VV && cv < VV) {
        const float s = n2[row] + n2[cv] - 2.0f * g[r];
        Dl[row * 32 + cv] = sqrtf(fmaxf(s, 1e-12f));
      }
    }
  };
  store_tile(g00, 0, 0);
  store_tile(g01, 0, 1);
  store_tile(g10, 1, 0);
  store_tile(g11, 1, 1);
  __syncthreads();

  // Local min/max over the 25x25 block, wave-reduce, global atomics.
  float lmin = INFINITY, lmax = 0.0f;
  if (lane < VV) {
    for (int j = 0; j < VV; ++j) {
      const float d = Dl[lane * 32 + j];
      lmin = fminf(lmin, d);
      lmax = fmaxf(lmax, d);
    }
  }
  for (int o = 16; o > 0; o >>= 1) {
    lmin = fminf(lmin, __shfl_xor(lmin, o, 32));
    lmax = fmaxf(lmax, __shfl_xor(lmax, o, 32));
  }
  if (lane == 0) {
    atomicMin(&mm[0], __float_as_uint(lmin));  // floats >= 0: bit-monotone
    atomicMax(&mm[1], __float_as_uint(lmax));
  }

  // Prim's MST (single-linkage deaths), lane j owns mind[j].
  bool vis = (lane == 0);
  float mind = (lane < VV && !vis) ? Dl[0 * 32 + lane] : INFINITY;
  float* dout = deaths + b * PP;

  for (int step = 0; step < PP; ++step) {
    // wave argmin with lowest-index tie-break (matches jnp.argmin)
    float bv = mind;
    int bj = lane;
    for (int o = 16; o > 0; o >>= 1) {
      const float ov = __shfl_xor(bv, o, 32);
      const int oj = __shfl_xor(bj, o, 32);
      if (ov < bv || (ov == bv && oj < bj)) {
        bv = ov;
        bj = oj;
      }
    }
    if (lane == 0) dout[step] = bv;  // raw death; normalized later
    if (lane == bj) vis = true;
    const float dj = (lane < VV) ? Dl[bj * 32 + lane] : INFINITY;
    mind = vis ? INFINITY : fminf(mind, dj);
  }
}

// ---------------------------------------------------------------------------
// Kernel 3: structure-element layer.
// out[b,e] = sum_p exp(-(s0^2*c0^2 + s1^2*(dn_p - c1)^2)),
// dn_p = (death_p - dmin) / (dmax - dmin)
// ---------------------------------------------------------------------------
__global__ void se_kernel(const float* __restrict__ centres,
                          const float* __restrict__ sharp,
                          const float* __restrict__ deaths,
                          const unsigned* __restrict__ mm,
                          float* __restrict__ out) {
  const int b = blockIdx.x;
  const int e = threadIdx.x;

  __shared__ float dn[PP];
  const float dmin = __uint_as_float(mm[0]);
  const float dmax = __uint_as_float(mm[1]);
  const float inv = 1.0f / (dmax - dmin);
  if (e < PP) dn[e] = (deaths[b * PP + e] - dmin) * inv;
  __syncthreads();

  const float c0 = centres[e * 2 + 0];
  const float c1 = centres[e * 2 + 1];
  const float s0 = sharp[e * 2 + 0];
  const float s1 = sharp[e * 2 + 1];
  const float t0 = s0 * c0;
  const float base = t0 * t0;  // s0^2 * (0 - c0)^2
  const float s1s = s1 * s1;

  float acc = 0.0f;
#pragma unroll
  for (int p = 0; p < PP; ++p) {
    const float dd = dn[p] - c1;
    acc += expf(-(base + s1s * dd * dd));
  }
  out[b * EE + e] = acc;
}

// ---------------------------------------------------------------------------
extern "C" void kernel_launch(void* const* d_in, const int* in_sizes, int n_in,
                              void* d_out, int out_size, void* d_ws,
                              size_t ws_size, hipStream_t stream) {
  (void)in_sizes; (void)n_in; (void)out_size; (void)ws_size;

  const float* x       = (const float*)d_in[0];  // [512,1,3,128,25]
  const float* centres = (const float*)d_in[1];  // [64,2]
  const float* sharp   = (const float*)d_in[2];  // [64,2]
  float* out = (float*)d_out;                    // [512,64]

  // Workspace layout (floats): xm | deaths | minmax(2 uints)
  float* xm = (float*)d_ws;                   // 512*3*25 = 38400 floats
  float* deaths = xm + NB * CC * VV;          // 512*24   = 12288 floats
  unsigned* mm = (unsigned*)(deaths + NB * PP);

  init_minmax_kernel<<<1, 32, 0, stream>>>(mm);
  mean_t_kernel<<<NB, 96, 0, stream>>>(x, xm);
  dist_mst_kernel<<<NB, 32, 0, stream>>>(xm, deaths, mm);
  se_kernel<<<NB, EE, 0, stream>>>(centres, sharp, deaths, mm, out);
}